// SRALNet_78847009620742
// MI455X (gfx1250) — compile-verified
//
#include <hip/hip_runtime.h>
#include <math.h>

#define N_  8
#define C_  128
#define K_  64
#define S1_ 5
#define M_  (K_*S1_)   /* 320 */
#define P_  1936       /* 44*44 = 121*16 */
#define ALPHA_ 1500.0f
#define EPS_ 1e-12f

typedef __attribute__((ext_vector_type(2))) float v2f;
typedef __attribute__((ext_vector_type(8))) float v8f;

__device__ __forceinline__ v8f wmma_f32_k4(v2f a, v2f b, v8f c) {
    // D = A(16x4 f32) * B(4x16 f32) + C(16x16 f32)
    return __builtin_amdgcn_wmma_f32_16x16x4_f32(
        /*neg_a=*/false, a, /*neg_b=*/false, b,
        /*c_mod=*/(short)0, c, /*reuse_a=*/false, /*reuse_b=*/false);
}

// ---------------- Kernel 0: bias vector b[m] = -ALPHA * ||cent_row_m|| ----------------
__global__ void sral_bias_kernel(const float* __restrict__ cent, float* __restrict__ bvec) {
    int m = threadIdx.x;
    if (m < M_) {
        const float* row = cent + m * C_;
        float s = 0.f;
        #pragma unroll 8
        for (int c = 0; c < C_; ++c) { float v = row[c]; s += v * v; }
        bvec[m] = -ALPHA_ * sqrtf(s);
    }
}

// ---------------- Kernel 1: logits GEMM (WMMA f32) + fused double softmax ----------------
// grid: (121 pixel-tiles, 8 batches); block: 320 threads = 10 waves.
// Each wave computes raw dot-product rows [wave*32, wave*32+32) for a 16-pixel tile;
// the 2*ALPHA scale and bias are folded into the epilogue fma (keeps the K-loop pure WMMA+loads).
__launch_bounds__(320)
__global__ void sral_logits_softmax_kernel(const float* __restrict__ x,
                                           const float* __restrict__ cent,
                                           const float* __restrict__ bvec,
                                           float* __restrict__ soft_assign) {
    __shared__ float ls[M_ * 16];     // 320x16 logits tile (20 KB)
    __shared__ float betas[K_ * 16];  // beta0 per (k, p)
    __shared__ float pmax[16];
    __shared__ float pinv[16];

    const int pt  = blockIdx.x;       // pixel tile, 0..120
    const int n   = blockIdx.y;
    const int p0  = pt * 16;
    const int tid = threadIdx.x;
    const int wave = tid >> 5;
    const int lane = tid & 31;
    const int hi = lane >> 4;         // lane half selects K sub-pair
    const int lm = lane & 15;

    const float* X = x + (size_t)n * C_ * P_;
    const int m0 = wave * 32;

    v8f acc0 = {}; v8f acc1 = {};
    for (int c0 = 0; c0 < C_; c0 += 4) {
        const int cb = c0 + 2 * hi;
        // B (4x16): row = channel cb+j, col = pixel lm  (p-contiguous across lanes)
        v2f b;
        b.x = X[(size_t)cb * P_ + p0 + lm];
        b.y = X[(size_t)(cb + 1) * P_ + p0 + lm];
        // A (16x4): raw centroid pair (contiguous -> b64 load), scale folded out of loop
        const float* a0p = cent + (size_t)(m0 + lm) * C_ + cb;
        v2f a0; a0.x = a0p[0]; a0.y = a0p[1];
        acc0 = wmma_f32_k4(a0, b, acc0);
        const float* a1p = cent + (size_t)(m0 + 16 + lm) * C_ + cb;
        v2f a1; a1.x = a1p[0]; a1.y = a1p[1];
        acc1 = wmma_f32_k4(a1, b, acc1);
    }
    // C layout: element (M = r + 8*hi, N = lm) in acc[r]; logits = 2*ALPHA*dot + b[m]
    #pragma unroll
    for (int r = 0; r < 8; ++r) {
        int mr = m0 + r + 8 * hi;
        ls[mr * 16 + lm] = fmaf(acc0[r], 2.f * ALPHA_, bvec[mr]);
        int mr1 = mr + 16;
        ls[mr1 * 16 + lm] = fmaf(acc1[r], 2.f * ALPHA_, bvec[mr1]);
    }
    __syncthreads();

    // beta softmax over S1 (5 values) at each (k, p); keep s=0 term
    for (int idx = tid; idx < K_ * 16; idx += 320) {
        int k = idx >> 4, p = idx & 15;
        float l[S1_];
        float mx = -1e30f;
        #pragma unroll
        for (int s = 0; s < S1_; ++s) { l[s] = ls[(k * S1_ + s) * 16 + p]; mx = fmaxf(mx, l[s]); }
        float se = 0.f;
        #pragma unroll
        for (int s = 0; s < S1_; ++s) se += __expf(l[s] - mx);
        betas[idx] = __expf(l[0] - mx) / se;
    }
    __syncthreads();

    // alpha softmax over K: per-pixel max & expsum by 16 threads...
    if (tid < 16) {
        const int p = tid;
        float mx = -1e30f;
        for (int k = 0; k < K_; ++k) mx = fmaxf(mx, ls[(k * S1_) * 16 + p]);
        float se = 0.f;
        for (int k = 0; k < K_; ++k) se += __expf(ls[(k * S1_) * 16 + p] - mx);
        pmax[p] = mx;
        pinv[p] = 1.f / se;
    }
    __syncthreads();
    // ...then all 320 threads combine and store (16-wide coalesced groups)
    {
        const size_t obase = (size_t)n * K_ * P_ + (size_t)p0;
        for (int idx = tid; idx < K_ * 16; idx += 320) {
            int k = idx >> 4, p = idx & 15;
            float alpha = __expf(ls[(k * S1_) * 16 + p] - pmax[p]) * pinv[p];
            soft_assign[obase + (size_t)k * P_ + p] = alpha * betas[idx];
        }
    }
}

// ---------------- Kernel 2: aggregation GEMM (WMMA f32) + rep subtract + norms ----------------
// grid: 8 (one block per batch); block: 256 threads = 8 waves, 4 tiles/wave -> 64x128 out.
// Uses sum_p (1+sa)*x = colsum + sum_p sa*x so the K-loop A operand is a bare b64 load.
__launch_bounds__(256)
__global__ void sral_aggregate_kernel(const float* __restrict__ x,
                                      const float* __restrict__ cent,
                                      const float* __restrict__ soft_assign,
                                      float* __restrict__ flat_out) {
    __shared__ float outs[K_ * C_];   // 8192 floats (32 KB)
    __shared__ float sums[K_];        // rowsum of w_assign per k
    __shared__ float cs[C_];          // colsum of x per channel
    __shared__ float part[256];
    __shared__ float part2[256];
    __shared__ float scale_s;

    const int n   = blockIdx.x;
    const int tid = threadIdx.x;
    const float* X  = x + (size_t)n * C_ * P_;
    const float* SA = soft_assign + (size_t)n * K_ * P_;

    // rowsum partials: 4 threads per k over sa
    {
        int k = tid >> 2, q = tid & 3;
        float s = 0.f;
        for (int p = q; p < P_; p += 4) s += SA[(size_t)k * P_ + p];
        part[tid] = s;
    }
    // colsum partials: 2 threads per channel over x
    {
        int c = tid >> 1, h = tid & 1;
        float s = 0.f;
        for (int p = h; p < P_; p += 2) s += X[(size_t)c * P_ + p];
        part2[tid] = s;
    }
    __syncthreads();
    if (tid < K_)
        sums[tid] = (float)P_ + part[4*tid] + part[4*tid+1] + part[4*tid+2] + part[4*tid+3];
    if (tid < C_)
        cs[tid] = part2[2*tid] + part2[2*tid+1];
    __syncthreads();

    const int wave = tid >> 5, lane = tid & 31;
    const int hi = lane >> 4, lm = lane & 15;
    const int tm  = wave >> 1;        // row tile 0..3  (m0 = tm*16)
    const int tcb = (wave & 1) * 4;   // column tile base (4 tiles of 16 channels)
    const int m0  = tm * 16;

    v8f acc[4];
    #pragma unroll
    for (int i = 0; i < 4; ++i) acc[i] = (v8f){};

    for (int p0 = 0; p0 < P_; p0 += 4) {
        const int pb = p0 + 2 * hi;
        // A (16x4): row k = m0+lm, K pair (pixels) contiguous -> b64 load, no VALU prep
        const float* ap = SA + (size_t)(m0 + lm) * P_ + pb;
        v2f a; a.x = ap[0]; a.y = ap[1];
        #pragma unroll
        for (int i = 0; i < 4; ++i) {
            const int c0 = (tcb + i) * 16;
            // B (4x16): row p = pb+j, col c = c0+lm; per-lane contiguous pair in x
            const float* bp = X + (size_t)(c0 + lm) * P_ + pb;
            v2f b; b.x = bp[0]; b.y = bp[1];
            acc[i] = wmma_f32_k4(a, b, acc[i]);
        }
    }
    #pragma unroll
    for (int i = 0; i < 4; ++i) {
        const int c0 = (tcb + i) * 16;
        #pragma unroll
        for (int r = 0; r < 8; ++r) {
            int k  = m0 + r + 8 * hi;
            int cc = c0 + lm;
            outs[k * C_ + cc] = acc[i][r] + cs[cc] - cent[(k * S1_) * C_ + cc] * sums[k];
        }
    }
    __syncthreads();

    // per-(n,k) L2 normalize over C
    if (tid < K_) {
        float ss = 0.f;
        #pragma unroll 8
        for (int c = 0; c < C_; ++c) { float v = outs[tid * C_ + c]; ss += v * v; }
        float rn = 1.f / fmaxf(sqrtf(ss), EPS_);
        for (int c = 0; c < C_; ++c) outs[tid * C_ + c] *= rn;
    }
    __syncthreads();

    // whole-vector L2 normalize over K*C
    {
        float s = 0.f;
        for (int idx = tid; idx < K_ * C_; idx += 256) { float v = outs[idx]; s += v * v; }
        part[tid] = s;
    }
    __syncthreads();
    if (tid == 0) {
        float t = 0.f;
        for (int i = 0; i < 256; ++i) t += part[i];
        scale_s = 1.f / fmaxf(sqrtf(t), EPS_);
    }
    __syncthreads();
    const float sc = scale_s;
    for (int idx = tid; idx < K_ * C_; idx += 256)
        flat_out[(size_t)n * K_ * C_ + idx] = outs[idx] * sc;
}

extern "C" void kernel_launch(void* const* d_in, const int* in_sizes, int n_in,
                              void* d_out, int out_size, void* d_ws, size_t ws_size,
                              hipStream_t stream) {
    (void)in_sizes; (void)n_in; (void)out_size; (void)ws_size;
    const float* x    = (const float*)d_in[0];   // (8,128,44,44) f32
    const float* cent = (const float*)d_in[1];   // (64,5,128) f32

    float* out  = (float*)d_out;
    float* flat = out;                              // N*K*C = 65536 floats
    float* sa   = out + (size_t)N_ * K_ * C_;       // soft_assign: N*K*1*P floats
    float* bvec = (float*)d_ws;                     // 320 floats of scratch

    sral_bias_kernel<<<1, M_, 0, stream>>>(cent, bvec);
    dim3 g1(P_ / 16, N_, 1);
    sral_logits_softmax_kernel<<<g1, 320, 0, stream>>>(x, cent, bvec, sa);
    sral_aggregate_kernel<<<N_, 256, 0, stream>>>(x, cent, sa, flat);
}